// UOTSolver_41944650612874
// MI455X (gfx1250) — compile-verified
//
#include <hip/hip_runtime.h>
#include <math.h>
#include <stdint.h>

#define Nn 131072
#define Kk 256
#define KE 257

static constexpr float EPS8 = 1e-8f;
static constexpr float TAU  = 0.95f;

// ---- workspace layout (4-byte units) ----
#define WS_T2    0          // 131072 floats
#define WS_S2    131072     // 256
#define WS_V     131328     // 131072
#define WS_U     262400     // 257
#define WS_KVP   262672     // 257*16 = 4112
#define WS_HIST  266784     // 256 u32
#define WS_PFX   267040
#define WS_MSK   267041
#define WS_KREM  267042
#define WS_Q0    267043
#define WS_Q1    267044
#define WS_DBC   267045
#define WS_LP    267056     // 4112
#define WS_TP    271168     // 4112
#define WS_DP    275280     // 4112  (ends ~279392 -> ~1.07 MB ws)

typedef float v2f __attribute__((ext_vector_type(2)));
typedef float v8f __attribute__((ext_vector_type(8)));

// ---------------- row norms (one WG per row of 256 floats) ----------------
__global__ void uot_rownorm(const float* __restrict__ X, float* __restrict__ out) {
  __shared__ float red[256];
  const int row = blockIdx.x, t = threadIdx.x;
  float x = X[(size_t)row * 256 + t];
  red[t] = x * x;
  __syncthreads();
  for (int s = 128; s > 0; s >>= 1) { if (t < s) red[t] += red[t + s]; __syncthreads(); }
  if (t == 0) out[row] = red[0];
}

// ---------------- cost GEMM via V_WMMA_F32_16X16X4_F32 ----------------
// WG(256 thr = 8 waves) computes a 16(k) x 128(n) tile of cost = sqrt(max(s2+t2-2*S@T^T,0))
__launch_bounds__(256)
__global__ void uot_gemm_cost(const float* __restrict__ S, const float* __restrict__ T,
                              const float* __restrict__ s2, const float* __restrict__ t2,
                              float* __restrict__ cost, float* __restrict__ cext) {
  __shared__ float St[16 * 68];    // stride 68 -> bank-conflict-free frag loads
  __shared__ float Tt[128 * 68];
  __shared__ float s2s[16];
  const int t    = threadIdx.x;
  const int k0   = blockIdx.x * 16;    // 16 k-bands
  const int n0   = blockIdx.y * 128;   // 1024 n-tiles
  if (t < 16) s2s[t] = s2[k0 + t];

  const int w    = t >> 5;             // wave 0..7 -> n sub-tile
  const int lane = t & 31;
  const int r    = lane & 15;
  const int chi  = lane >> 4;          // K-pair select (ISA 16x4 / 4x16 f32 layout)
  v8f acc = {0.f, 0.f, 0.f, 0.f, 0.f, 0.f, 0.f, 0.f};

  for (int d0 = 0; d0 < 256; d0 += 64) {
    __syncthreads();
    for (int idx = t; idx < 16 * 64; idx += 256) {          // S chunk 16x64
      int rr = idx >> 6, cc = idx & 63;
      St[rr * 68 + cc] = S[(size_t)(k0 + rr) * 256 + d0 + cc];
    }
    for (int idx = t; idx < 128 * 16; idx += 256) {         // T chunk 128x64 (float4)
      int rr = idx >> 4, q = idx & 15;
      const float4 val = *(const float4*)&T[(size_t)(n0 + rr) * 256 + d0 + q * 4];
      float* dst = &Tt[rr * 68 + q * 4];
      dst[0] = val.x; dst[1] = val.y; dst[2] = val.z; dst[3] = val.w;
    }
    __syncthreads();
    for (int dd = 0; dd < 64; dd += 4) {
      const int c = dd + 2 * chi;
      v2f a, b;
      a[0] = St[r * 68 + c];               a[1] = St[r * 68 + c + 1];
      b[0] = Tt[(w * 16 + r) * 68 + c];    b[1] = Tt[(w * 16 + r) * 68 + c + 1];
      acc = __builtin_amdgcn_wmma_f32_16x16x4_f32(false, a, false, b, (short)0, acc,
                                                  false, false);
    }
  }
  const int   n   = n0 + w * 16 + r;   // C/D layout: N = lane&15
  const float t2n = t2[n];
  for (int rr = 0; rr < 8; ++rr) {     // M = rr + 8*(lane>>4)
    const int   k  = k0 + rr + 8 * chi;
    const float g  = acc[rr];
    const float d2 = s2s[rr + 8 * chi] + t2n - 2.0f * g;
    const float cv = sqrtf(fmaxf(d2, 0.0f));
    cost[(size_t)k * Nn + n] = cv;
    cext[(size_t)k * Nn + n] = cv;
  }
}

// ---------------- per-column min over 256 rows ----------------
__global__ void uot_colmin(const float* __restrict__ cost, float* __restrict__ mind) {
  const int j = blockIdx.x * blockDim.x + threadIdx.x;
  float m = 3.4e38f;
  for (int k = 0; k < Kk; ++k) m = fminf(m, cost[(size_t)k * Nn + j]);
  mind[j] = m;
}

// ---------------- radix-select (quantile) ----------------
__global__ void uot_selinit(unsigned* wsu, unsigned krem) {
  const int t = threadIdx.x;
  if (t < 256) wsu[WS_HIST + t] = 0u;
  if (t == 0) { wsu[WS_PFX] = 0u; wsu[WS_MSK] = 0u; wsu[WS_KREM] = krem; }
}

__global__ void uot_hist(const float* __restrict__ cost, unsigned* wsu, int shift) {
  __shared__ unsigned h[256];
  const int t = threadIdx.x;
  h[t] = 0u;
  __syncthreads();
  const unsigned prefix = wsu[WS_PFX], mask = wsu[WS_MSK];
  const size_t   stride = (size_t)gridDim.x * 256;
  for (size_t idx = (size_t)blockIdx.x * 256 + t; idx < (size_t)Kk * Nn; idx += stride) {
    const unsigned key = __float_as_uint(cost[idx]);   // cost>=0 -> uint order == float order
    if ((key & mask) == prefix) atomicAdd(&h[(key >> shift) & 255u], 1u);
  }
  __syncthreads();
  if (h[t]) atomicAdd(&wsu[WS_HIST + t], h[t]);
}

__global__ void uot_scan(unsigned* wsu, int shift, int qslot) {
  if (threadIdx.x != 0) return;
  unsigned k = wsu[WS_KREM], running = 0, chosen = 0;
  for (int b = 0; b < 256; ++b) {
    const unsigned c = wsu[WS_HIST + b];
    if (k < running + c) { chosen = (unsigned)b; break; }
    running += c;
  }
  k -= running;
  const unsigned prefix = wsu[WS_PFX] | (chosen << shift);
  wsu[WS_PFX] = prefix;
  wsu[WS_MSK] |= (0xFFu << shift);
  wsu[WS_KREM] = k;
  for (int b = 0; b < 256; ++b) wsu[WS_HIST + b] = 0u;
  if (shift == 0) wsu[qslot] = prefix;
}

__global__ void uot_dustbin(const unsigned* wsu, float* wsf, float* out_dbc) {
  if (threadIdx.x == 0) {
    const float q0 = __uint_as_float(wsu[WS_Q0]);
    const float q1 = __uint_as_float(wsu[WS_Q1]);
    const float db = 0.2f * q0 + 0.8f * q1;   // frac(0.8*(M-1)) = 0.8
    wsf[WS_DBC] = db;
    *out_dbc    = db;
  }
}

// ---------------- dustbin row + distance_unknown_score + v init ----------------
__global__ void uot_fillrow(const float* wsf, float* __restrict__ cext,
                            const float* __restrict__ mind,
                            float* __restrict__ dus, float* __restrict__ v) {
  const int   j  = blockIdx.x * blockDim.x + threadIdx.x;
  const float db = wsf[WS_DBC];
  cext[(size_t)Kk * Nn + j] = db;
  const float x = (mind[j] - db) * 20.0f;     // /EPSILON
  dus[j] = 1.0f / (1.0f + expf(-x));
  v[j]   = 1.0f;
}

// ---------------- kernel matrix into plan region ----------------
__global__ void uot_kmat(const float* __restrict__ cext, float* __restrict__ km) {
  const size_t idx = (size_t)blockIdx.x * blockDim.x + threadIdx.x;
  if (idx < (size_t)KE * Nn) km[idx] = fmaxf(expf(-cext[idx] * 20.0f), EPS8);
}

// ---------------- Sinkhorn: kv partials (row pass) ----------------
__global__ void uot_kvpart(const float* __restrict__ km, const float* __restrict__ v,
                           float* __restrict__ kvp) {
  __shared__ float red[256];
  const int i = blockIdx.x, c = blockIdx.y, t = threadIdx.x;
  const float* row = km + (size_t)i * Nn + (size_t)c * 8192;
  const float* vv  = v + (size_t)c * 8192;
  float acc = 0.f;
  for (int s = 0; s < 32; ++s) { const int j = t + s * 256; acc += row[j] * vv[j]; }
  red[t] = acc;
  __syncthreads();
  for (int s = 128; s > 0; s >>= 1) { if (t < s) red[t] += red[t + s]; __syncthreads(); }
  if (t == 0) kvp[i * 16 + c] = red[0];
}

__global__ void uot_uupd(const float* __restrict__ kvp, float* __restrict__ u) {
  const int i = threadIdx.x;
  if (i < KE) {
    float s = 0.f;
    for (int c = 0; c < 16; ++c) s += kvp[i * 16 + c];
    const float kv = fmaxf(s, EPS8);
    const float sm = (i == 256) ? 0.05f : (0.95f / 256.0f);
    u[i] = powf(sm / kv, TAU);
  }
}

// ---------------- Sinkhorn: column pass (ktu fused with v update) ----------------
__global__ void uot_vupd(const float* __restrict__ km, const float* __restrict__ u,
                         float* __restrict__ v) {
  __shared__ float su[KE];
  const int t = threadIdx.x;
  for (int i = t; i < KE; i += 256) su[i] = u[i];
  __syncthreads();
  const int j = blockIdx.x * 256 + t;
  float acc = 0.f;
  for (int i = 0; i < KE; ++i) acc += km[(size_t)i * Nn + j] * su[i];
  const float kt = fmaxf(acc, EPS8);
  v[j] = powf((1.0f / (float)Nn) / kt, TAU);
}

// ---------------- plan = u*K*v in place + deterministic partial sums ----------------
__global__ void uot_planfin(float* __restrict__ plan, const float* __restrict__ cext,
                            const float* __restrict__ u, const float* __restrict__ v,
                            float* __restrict__ lp, float* __restrict__ tp,
                            float* __restrict__ dp) {
  __shared__ float r1[256], r2[256], r3[256];
  const int    t    = threadIdx.x;
  const size_t base = (size_t)blockIdx.x * 8192;
  float ls = 0.f, ts = 0.f, ds = 0.f;
  for (int s = 0; s < 32; ++s) {
    const size_t idx = base + t + s * 256;
    const int i = (int)(idx >> 17);
    const int j = (int)(idx & (Nn - 1));
    const float p = u[i] * plan[idx] * v[j];
    plan[idx] = p;
    ls += p * cext[idx];
    ts += p;
    if (i == 256) ds += p;
  }
  r1[t] = ls; r2[t] = ts; r3[t] = ds;
  __syncthreads();
  for (int s = 128; s > 0; s >>= 1) {
    if (t < s) { r1[t] += r1[t + s]; r2[t] += r2[t + s]; r3[t] += r3[t + s]; }
    __syncthreads();
  }
  if (t == 0) { lp[blockIdx.x] = r1[0]; tp[blockIdx.x] = r2[0]; dp[blockIdx.x] = r3[0]; }
}

__global__ void uot_scalars(const float* __restrict__ lp, const float* __restrict__ tp,
                            const float* __restrict__ dp,
                            float* out_loss, float* out_ratio) {
  __shared__ float r1[256], r2[256], r3[256];
  const int t = threadIdx.x;
  float a = 0.f, b = 0.f, c = 0.f;
  for (int m = t; m < 4112; m += 256) { a += lp[m]; b += tp[m]; c += dp[m]; }
  r1[t] = a; r2[t] = b; r3[t] = c;
  __syncthreads();
  for (int s = 128; s > 0; s >>= 1) {
    if (t < s) { r1[t] += r1[t + s]; r2[t] += r2[t + s]; r3[t] += r3[t + s]; }
    __syncthreads();
  }
  if (t == 0) { *out_loss = r1[0]; *out_ratio = r3[0] / fmaxf(r2[0], EPS8); }
}

// ---------------- per-column outputs (incl. class_scores transpose) ----------------
__launch_bounds__(256)
__global__ void uot_colout(const float* __restrict__ plan,
                           float* __restrict__ cs, float* __restrict__ dsc,
                           float* __restrict__ bkm, float* __restrict__ mus,
                           float* __restrict__ asg) {
  __shared__ float tile[KE * 32];
  __shared__ float invden[32];
  const int t  = threadIdx.x;
  const int j0 = blockIdx.x * 32;
  for (int idx = t; idx < KE * 32; idx += 256) {
    const int i = idx >> 5, c = idx & 31;
    tile[idx] = plan[(size_t)i * Nn + j0 + c];
  }
  __syncthreads();
  if (t < 32) {
    const int c = t;
    float known = 0.f, best = -1.f;
    int   amax  = 0;
    for (int i = 0; i < Kk; ++i) {
      const float val = tile[i * 32 + c];
      known += val;
      if (val > best) { best = val; amax = i; }   // first occurrence of max
    }
    const float last   = tile[256 * 32 + c];
    const float colsum = known + last;
    const float denom  = fmaxf(known, EPS8);
    const float bk     = best / denom;
    const float dv     = last / fmaxf(colsum, EPS8);
    const int   j      = j0 + c;
    bkm[j] = bk;
    dsc[j] = dv;
    mus[j] = dv / (dv + bk + EPS8);
    asg[j] = (float)amax;
    invden[c] = 1.0f / denom;
  }
  __syncthreads();
  const int c = t >> 3, l = t & 7;
  const float inv = invden[c];
  float* dst = cs + (size_t)(j0 + c) * 256;
  for (int s = 0; s < 32; ++s) {
    const int k = s * 8 + l;
    dst[k] = tile[k * 32 + c] * inv;
  }
}

extern "C" void kernel_launch(void* const* d_in, const int* in_sizes, int n_in,
                              void* d_out, int out_size, void* d_ws, size_t ws_size,
                              hipStream_t stream) {
  (void)in_sizes; (void)n_in; (void)out_size; (void)ws_size;
  const float* S = (const float*)d_in[0];   // source_proto [256,256]
  const float* T = (const float*)d_in[1];   // target_feat  [131072,256]
  float*    out = (float*)d_out;
  float*    wsf = (float*)d_ws;
  unsigned* wsu = (unsigned*)d_ws;

  float* o_cost = out;
  float* o_cext = out + (size_t)Kk * Nn;
  float* o_mind = o_cext + (size_t)KE * Nn;
  float* o_plan = o_mind + Nn;
  float* o_loss = o_plan + (size_t)KE * Nn;
  float* o_dbc  = o_loss + 1;
  float* o_rat  = o_dbc + 1;
  float* o_dus  = o_rat + 1;
  float* o_cs   = o_dus + Nn;
  float* o_dsc  = o_cs + (size_t)Nn * Kk;
  float* o_bkm  = o_dsc + Nn;
  float* o_mus  = o_bkm + Nn;
  float* o_asg  = o_mus + Nn;

  float* t2  = wsf + WS_T2;
  float* s2  = wsf + WS_S2;
  float* v   = wsf + WS_V;
  float* u   = wsf + WS_U;
  float* kvp = wsf + WS_KVP;
  float* lp  = wsf + WS_LP;
  float* tp  = wsf + WS_TP;
  float* dp  = wsf + WS_DP;

  uot_rownorm<<<Nn, 256, 0, stream>>>(T, t2);
  uot_rownorm<<<Kk, 256, 0, stream>>>(S, s2);
  uot_gemm_cost<<<dim3(16, 1024), 256, 0, stream>>>(S, T, s2, t2, o_cost, o_cext);
  uot_colmin<<<Nn / 256, 256, 0, stream>>>(o_cost, o_mind);

  // exact order statistics k0/k1 for q=0.8 of 33,554,432 values
  const unsigned k0 = 26843544u, k1 = 26843545u;
  for (int which = 0; which < 2; ++which) {
    uot_selinit<<<1, 256, 0, stream>>>(wsu, which ? k1 : k0);
    for (int shift = 24; shift >= 0; shift -= 8) {
      uot_hist<<<1024, 256, 0, stream>>>(o_cost, wsu, shift);
      uot_scan<<<1, 64, 0, stream>>>(wsu, shift, which ? WS_Q1 : WS_Q0);
    }
  }
  uot_dustbin<<<1, 64, 0, stream>>>(wsu, wsf, o_dbc);
  uot_fillrow<<<Nn / 256, 256, 0, stream>>>(wsf, o_cext, o_mind, o_dus, v);
  uot_kmat<<<(KE * Nn) / 256, 256, 0, stream>>>(o_cext, o_plan);  // K matrix lives in plan slot

  for (int it = 0; it < 30; ++it) {
    uot_kvpart<<<dim3(KE, 16), 256, 0, stream>>>(o_plan, v, kvp);
    uot_uupd<<<1, 288, 0, stream>>>(kvp, u);
    uot_vupd<<<Nn / 256, 256, 0, stream>>>(o_plan, u, v);
  }

  uot_planfin<<<4112, 256, 0, stream>>>(o_plan, o_cext, u, v, lp, tp, dp);
  uot_scalars<<<1, 256, 0, stream>>>(lp, tp, dp, o_loss, o_rat);
  uot_colout<<<Nn / 32, 256, 0, stream>>>(o_plan, o_cs, o_dsc, o_bkm, o_mus, o_asg);
}